// PCEN_83279415869875
// MI455X (gfx1250) — compile-verified
//
#include <hip/hip_runtime.h>

// PCEN for x:(B,F,T=8000) f32, log_s:(F,) f32.
// One workgroup per (b,f) channel: async-copy row to LDS (skewed),
// affine-map parallel scan for the IIR, pointwise transform, async-copy back.

#define TLEN     8000
#define NTHREADS 256
#define CHUNK    32
#define NACT     250      // TLEN / CHUNK (exact)
#define NVEC     2000     // TLEN / 4 (float4 packets)
#define NWAVE    8

// skewed LDS word index: +1 word per 32 elements -> stride 33, coprime with 64 banks
__device__ __forceinline__ unsigned skw(unsigned t) { return t + (t >> 5); }

__device__ __forceinline__ unsigned lds_off_of(const void* p) {
  // addrspace(3) offset == low 32 bits of the flat address for LDS aperture
  return (unsigned)(unsigned long long)p;
}

__global__ __launch_bounds__(NTHREADS)
void PCEN_83279415869875_kernel(const float* __restrict__ x,
                                const float* __restrict__ log_s,
                                float* __restrict__ out, int F) {
  __shared__ float xs[TLEN + NACT];   // skewed storage: 33000 B
  __shared__ float wA[NWAVE], wB[NWAVE];

  const int c   = blockIdx.x;            // channel = b*F + f
  const int f   = c % F;
  const float* gx = x   + (size_t)c * TLEN;
  float*       gy = out + (size_t)c * TLEN;

  const int tid  = threadIdx.x;
  const int lane = tid & 31;
  const int wid  = tid >> 5;

  const float s = __builtin_amdgcn_exp2f(log_s[f] * 1.4426950408889634f); // exp(log_s)
  const float a = 1.0f - s;

  // ---------- async global -> LDS (B128 per lane), skewed ----------
  const unsigned base_lds = lds_off_of(xs);
  #pragma unroll
  for (int r = 0; r < 8; ++r) {
    unsigned idx = (unsigned)tid + (unsigned)r * NTHREADS;
    if (idx < NVEC) {
      unsigned go = idx * 16u;
      unsigned lo = base_lds + idx * 16u + ((idx >> 3) << 2); // +4B per 8 packets
      unsigned long long sb = (unsigned long long)gx;
      asm volatile("global_load_async_to_lds_b128 %0, %1, %2"
                   :: "v"(lo), "v"(go), "s"(sb) : "memory");
    }
  }
  asm volatile("s_wait_asynccnt 0" ::: "memory");
  __syncthreads();

  const float m0 = xs[0];  // h[-1] = x[0]; uniform recurrence then gives M[0]=x[0]

  // ---------- pass 1: per-chunk affine map (A,B): m_out = A*m_in + B ----------
  float Av = 1.0f, Bv = 0.0f;
  if (tid < NACT) {
    float b = 0.0f;
    const int t0 = tid * CHUNK;
    #pragma unroll
    for (int j = 0; j < CHUNK; ++j) {
      float xv = xs[skw(t0 + j)];
      b = fmaf(a, b, s * xv);
    }
    float a2 = a * a, a4 = a2 * a2, a8 = a4 * a4, a16 = a8 * a8;
    Av = a16 * a16;   // a^32
    Bv = b;
  }

  // ---------- intra-wave inclusive scan of affine composition ----------
  #pragma unroll
  for (int off = 1; off < 32; off <<= 1) {
    float Ap = __shfl_up(Av, off, 32);
    float Bp = __shfl_up(Bv, off, 32);
    if (lane >= off) {
      Bv = fmaf(Av, Bp, Bv);   // earlier map first, then mine
      Av = Av * Ap;
    }
  }
  if (lane == 31) { wA[wid] = Av; wB[wid] = Bv; }
  __syncthreads();

  // exclusive cross-wave prefix (<=7 composes, trivially cheap)
  float Aw = 1.0f, Bw = 0.0f;
  for (int w = 0; w < wid; ++w) {
    Bw = fmaf(wA[w], Bw, wB[w]);
    Aw = Aw * wA[w];
  }
  // exclusive intra-wave prefix
  float Ae = __shfl_up(Av, 1, 32);
  float Be = __shfl_up(Bv, 1, 32);
  if (lane == 0) { Ae = 1.0f; Be = 0.0f; }
  // carry-in for this chunk: (wave_excl then lane_excl) applied to m0
  float At = Aw * Ae;
  float Bt = fmaf(Ae, Bw, Be);
  float m  = fmaf(At, m0, Bt);

  // ---------- pass 2: replay IIR + pointwise PCEN, write back to LDS in place ----------
  if (tid < NACT) {
    const int t0 = tid * CHUNK;
    #pragma unroll
    for (int j = 0; j < CHUNK; ++j) {
      unsigned w  = skw(t0 + j);
      float    xv = xs[w];
      m = fmaf(a, m, s * xv);                               // m == M[t]
      float lg = __builtin_amdgcn_logf(m + 1e-6f);          // log2(eps+M)
      float p  = __builtin_amdgcn_exp2f(-0.98f * lg);       // (eps+M)^-alpha
      float v  = fmaf(xv, p, 2.0f);                         // x/(eps+M)^a + delta
      xs[w] = __builtin_amdgcn_sqrtf(v) - 1.4142135623730951f; // ^0.5 - delta^0.5
    }
  }
  __syncthreads();

  // ---------- async LDS -> global (coalesced B128) ----------
  #pragma unroll
  for (int r = 0; r < 8; ++r) {
    unsigned idx = (unsigned)tid + (unsigned)r * NTHREADS;
    if (idx < NVEC) {
      unsigned go = idx * 16u;
      unsigned lo = base_lds + idx * 16u + ((idx >> 3) << 2);
      unsigned long long sb = (unsigned long long)gy;
      asm volatile("global_store_async_from_lds_b128 %0, %1, %2"
                   :: "v"(go), "v"(lo), "s"(sb) : "memory");
    }
  }
  asm volatile("s_wait_asynccnt 0" ::: "memory");
}

extern "C" void kernel_launch(void* const* d_in, const int* in_sizes, int n_in,
                              void* d_out, int out_size, void* d_ws, size_t ws_size,
                              hipStream_t stream) {
  (void)n_in; (void)out_size; (void)d_ws; (void)ws_size;
  const float* x     = (const float*)d_in[0];
  const float* log_s = (const float*)d_in[1];
  float*       out   = (float*)d_out;
  const int F = in_sizes[1];                 // 128
  const int C = in_sizes[0] / TLEN;          // B*F = 4096 channels
  PCEN_83279415869875_kernel<<<C, NTHREADS, 0, stream>>>(x, log_s, out, F);
}